// Att_2_layer2_11879879544271
// MI455X (gfx1250) — compile-verified
//
#include <hip/hip_runtime.h>

typedef __attribute__((ext_vector_type(2))) float v2f;
typedef __attribute__((ext_vector_type(8))) float v8f;

#define B_  16
#define N_  128
#define A_  32
#define O_  64
#define VD_ 768
#define QD_ 768
#define H_  1024

__device__ __forceinline__ v8f wmma_f32(v2f a, v2f b, v8f c) {
  // D = A(16x4) x B(4x16) + C(16x16), exact fp32
  return __builtin_amdgcn_wmma_f32_16x16x4_f32(false, a, false, b, (short)0, c,
                                               false, false);
}

// ---------------------------------------------------------------------------
// Kernel 1: M[b] = obj_reps[b] (64x768) @ vw^T (768x1024)  -> [16][64][1024]
// Each wave computes a 16x64 tile of one batch's M.
// waves = 16 b * 4 mtiles * 16 hchunks = 1024  -> 128 blocks of 256 threads
// ---------------------------------------------------------------------------
__global__ __launch_bounds__(256) void k_objvw(const float* __restrict__ obj,
                                               const float* __restrict__ vw,
                                               float* __restrict__ Mout) {
  const int wave   = (blockIdx.x * blockDim.x + threadIdx.x) >> 5;
  const int lane   = threadIdx.x & 31;
  const int hchunk = wave & 15;        // 16 chunks of 64 columns
  const int mt     = (wave >> 4) & 3;  // 4 tiles of 16 rows
  const int b      = wave >> 6;        // 16 batches
  const int lo = lane & 15, hi = lane >> 4;

  const float* arow = obj + ((size_t)b * 64 + mt * 16 + lo) * VD_ + hi * 2;
  const int h0 = hchunk * 64;

  v8f acc[4] = {};
  for (int k = 0; k < VD_; k += 4) {
    v2f af = *(const v2f*)(arow + k);   // A[row][k..k+1] (lo half) / k+2..k+3 (hi half)
#pragma unroll
    for (int c = 0; c < 4; ++c) {
      // B[k'][h] = vw[h][k'] ; lanes 0-15: k,k+1 ; lanes 16-31: k+2,k+3
      v2f bf = *(const v2f*)(vw + (size_t)(h0 + c * 16 + lo) * VD_ + k + hi * 2);
      acc[c] = wmma_f32(af, bf, acc[c]);
    }
  }
#pragma unroll
  for (int c = 0; c < 4; ++c) {
#pragma unroll
    for (int r = 0; r < 8; ++r) {
      const int row = mt * 16 + hi * 8 + r;   // C layout: vgpr r -> M=r / M=8+r
      const int h   = h0 + c * 16 + lo;
      Mout[(size_t)b * (O_ * H_) + (size_t)row * H_ + h] = acc[c][r];
    }
  }
}

// ---------------------------------------------------------------------------
// Kernel 2: w2[bn][h] = relu(q[bn] . qw[h] + qb[h]) * lw[h]   -> [2048][1024]
// waves = 128 mtiles * 16 hchunks = 2048  -> 256 blocks of 256 threads
// ---------------------------------------------------------------------------
__global__ __launch_bounds__(256) void k_qproj(const float* __restrict__ q,
                                               const float* __restrict__ qw,
                                               const float* __restrict__ qb,
                                               const float* __restrict__ lw,
                                               float* __restrict__ w2) {
  const int wave   = (blockIdx.x * blockDim.x + threadIdx.x) >> 5;
  const int lane   = threadIdx.x & 31;
  const int hchunk = wave & 15;
  const int mt     = wave >> 4;        // 0..127 (2048 rows / 16)
  const int lo = lane & 15, hi = lane >> 4;

  const float* arow = q + (size_t)(mt * 16 + lo) * QD_ + hi * 2;
  const int h0 = hchunk * 64;

  v8f acc[4] = {};
  for (int k = 0; k < QD_; k += 4) {
    v2f af = *(const v2f*)(arow + k);
#pragma unroll
    for (int c = 0; c < 4; ++c) {
      v2f bf = *(const v2f*)(qw + (size_t)(h0 + c * 16 + lo) * QD_ + k + hi * 2);
      acc[c] = wmma_f32(af, bf, acc[c]);
    }
  }
#pragma unroll
  for (int c = 0; c < 4; ++c) {
    const int h = h0 + c * 16 + lo;
    const float bias = qb[h];
    const float scale = lw[h];
#pragma unroll
    for (int r = 0; r < 8; ++r) {
      const int row = mt * 16 + hi * 8 + r;
      float x = acc[c][r] + bias;
      x = x > 0.f ? x : 0.f;
      w2[(size_t)row * H_ + h] = x * scale;
    }
  }
}

// ---------------------------------------------------------------------------
// Kernel 3: per (b,n): P = att1_bn (32x64) @ M[b] (64x1024),
//   logits[a] = sum_h relu(P[a,h]+vb[h]) * w2[bn,h]  (+lb, /t, mask, softmax)
//   out[o] = sum_a att2[a] * att1_bn[a,o]
// One 256-thread block (8 waves) per (b,n); wave w owns h-columns [w*128, w*128+128)
// ---------------------------------------------------------------------------
__global__ __launch_bounds__(256) void k_att(const float* __restrict__ att1,
                                             const float* __restrict__ Mmat,
                                             const float* __restrict__ w2,
                                             const float* __restrict__ vb,
                                             const int*   __restrict__ tags,
                                             const int*   __restrict__ tptr,
                                             const float* __restrict__ lb,
                                             float* __restrict__ out) {
  const int bn = blockIdx.x;
  const int b  = bn >> 7;
  const int tid = threadIdx.x;
  const int lane = tid & 31, wv = tid >> 5;
  const int lo = lane & 15, hi = lane >> 4;

  __shared__ __align__(16) float att_lds[A_ * 68];  // 32 rows, stride 68 (bank-safe)
  __shared__ float logits_lds[A_];
  __shared__ float att2_lds[A_];

  // stage att1[b,n] (32x64 = 8 KB) into LDS, coalesced
  const float* a1 = att1 + (size_t)bn * (A_ * O_);
  for (int i = tid; i < A_ * O_; i += 256) {
    att_lds[(i >> 6) * 68 + (i & 63)] = a1[i];
  }
  if (tid < A_) logits_lds[tid] = 0.0f;
  __syncthreads();

  const float* Mb    = Mmat + (size_t)b * (O_ * H_);
  const float* w2row = w2 + (size_t)bn * H_;

  float lacc[16];
#pragma unroll
  for (int i = 0; i < 16; ++i) lacc[i] = 0.f;

  for (int c = 0; c < 8; ++c) {
    const int h = wv * 128 + c * 16 + lo;  // this lane's output column
    const float w2v = w2row[h];
    const float vbv = vb[h];
    v8f acc0 = {}, acc1 = {};
#pragma unroll
    for (int k = 0; k < 16; ++k) {
      const int kk = 4 * k + hi * 2;
      v2f af0 = *(const v2f*)(&att_lds[lo * 68 + kk]);         // rows 0-15
      v2f af1 = *(const v2f*)(&att_lds[(16 + lo) * 68 + kk]);  // rows 16-31
      const float* bp = Mb + (size_t)kk * H_ + h;
      v2f bf;
      bf.x = bp[0];    // B[kk][h]
      bf.y = bp[H_];   // B[kk+1][h]
      acc0 = wmma_f32(af0, bf, acc0);
      acc1 = wmma_f32(af1, bf, acc1);
    }
#pragma unroll
    for (int r = 0; r < 8; ++r) {
      float x0 = acc0[r] + vbv; x0 = x0 > 0.f ? x0 : 0.f;
      float x1 = acc1[r] + vbv; x1 = x1 > 0.f ? x1 : 0.f;
      lacc[r]     += x0 * w2v;   // rows (hi*8 + r)
      lacc[8 + r] += x1 * w2v;   // rows (16 + hi*8 + r)
    }
  }
  // reduce over the 16 lanes of each half-wave (columns of this wave's chunk)
#pragma unroll
  for (int m = 1; m <= 8; m <<= 1) {
#pragma unroll
    for (int i = 0; i < 16; ++i) lacc[i] += __shfl_xor(lacc[i], m, 32);
  }
  // one LDS float-atomic per lane (distributed across slots -> ds_add_f32)
#pragma unroll
  for (int i = 0; i < 16; ++i) {
    if (lo == i) {
      const int row = (i < 8) ? (hi * 8 + i) : (16 + hi * 8 + (i - 8));
      atomicAdd(&logits_lds[row], lacc[i]);
    }
  }
  __syncthreads();

  // masked softmax over a (wave 0, one lane per a)
  if (wv == 0) {
    const float t = (float)tptr[0];
    float lg = (logits_lds[lane] + lb[0]) / t;
    const bool ok = tags[(size_t)bn * A_ + lane] > 0;
    float ml = ok ? lg : -1e30f;
    float mx = ml;
#pragma unroll
    for (int m = 1; m <= 16; m <<= 1) {
      float o = __shfl_xor(mx, m, 32);
      mx = o > mx ? o : mx;
    }
    float e = __expf(ml - mx);
    float s = e;
#pragma unroll
    for (int m = 1; m <= 16; m <<= 1) s += __shfl_xor(s, m, 32);
    att2_lds[lane] = e / s;
  }
  __syncthreads();

  // out[bn][o] = sum_a att2[a] * att1[a][o]
  if (tid < O_) {
    float s = 0.f;
#pragma unroll
    for (int a = 0; a < A_; ++a) s += att2_lds[a] * att_lds[a * 68 + tid];
    out[(size_t)bn * O_ + tid] = s;
  }
}

extern "C" void kernel_launch(void* const* d_in, const int* in_sizes, int n_in,
                              void* d_out, int out_size, void* d_ws, size_t ws_size,
                              hipStream_t stream) {
  const float* q    = (const float*)d_in[0];
  const float* att1 = (const float*)d_in[1];
  const float* obj  = (const float*)d_in[2];
  const int*   tags = (const int*)d_in[3];
  const int*   t    = (const int*)d_in[4];
  const float* vw   = (const float*)d_in[5];
  const float* vb   = (const float*)d_in[6];
  const float* qw   = (const float*)d_in[7];
  const float* qb   = (const float*)d_in[8];
  const float* lw   = (const float*)d_in[9];
  const float* lb   = (const float*)d_in[10];

  float* Mws = (float*)d_ws;                            // 16*64*1024 f32 = 4 MB
  float* w2  = Mws + (size_t)B_ * O_ * H_;              // 2048*1024 f32 = 8 MB

  k_objvw<<<128, 256, 0, stream>>>(obj, vw, Mws);
  k_qproj<<<256, 256, 0, stream>>>(q, qw, qb, lw, w2);
  k_att<<<B_ * N_, 256, 0, stream>>>(att1, Mws, w2, vb, tags, t, lb,
                                     (float*)d_out);
}